// GNN_29600914604721
// MI455X (gfx1250) — compile-verified
//
#include <hip/hip_runtime.h>
#include <hip/hip_bf16.h>

// TAGConv GNN for MI455X (gfx1250, wave32).
// Dense hop-term GEMMs: V_WMMA_F32_16X16X4_F32 (full f32 precision).
// Sparse propagation: float4 gather (global_load_b128) + hardware
// global_atomic_add_f32 (inline asm, guaranteed no CAS fallback).
// Feature matrix (12.8 MB) is L2-resident (192 MB L2), so the 4.8 GB of
// hop traffic is served at L2 bandwidth, HBM only sees streaming passes.

typedef __attribute__((ext_vector_type(2))) float v2f;
typedef __attribute__((ext_vector_type(8))) float v8f;

#define NEG_SLOPE 0.01f

// Hardware no-return f32 atomic add (GLOBAL_ATOMIC_ADD_F32, STOREcnt-tracked;
// implicit wait at s_endpgm).
__device__ __forceinline__ void atomic_fadd_hw(float* p, float v) {
  asm volatile("global_atomic_add_f32 %0, %1, off" ::"v"(p), "v"(v) : "memory");
}

// ---------------------------------------------------------------------------
// GEMM: acc[N x Fout] (+)= H[N x FIN] @ W[FIN x Fout], optional bias + lrelu.
// One wave computes one 16x16 tile via WMMA f32 16x16x4.
// FIN is 32 or 64; N multiple of 16 (50000 = 3125*16).
// ---------------------------------------------------------------------------
template <int FIN, bool BETA, bool BIASRELU>
__global__ void tag_gemm_wmma(const float* __restrict__ H,
                              const float* __restrict__ W,
                              float* __restrict__ acc,
                              const float* __restrict__ bias,
                              int nNodes, int Fout) {
  const int lane = threadIdx.x & 31;
  const int wave = blockIdx.x * (blockDim.x >> 5) + (threadIdx.x >> 5);
  const int ntiles = Fout >> 4;
  const int mtiles = nNodes >> 4;
  if (wave >= mtiles * ntiles) return;  // wave-uniform: EXEC all-1s for WMMA

  const int mt = wave / ntiles;
  const int nt = wave - mt * ntiles;
  const int mbase = mt << 4;
  const int nbase = nt << 4;

  // A 16x4 f32: lanes 0-15 -> M=lane (K=k0,k0+1); lanes 16-31 -> M=lane-16 (K=k0+2,k0+3)
  const int arow = mbase + (lane & 15);
  const int khalf = (lane >> 4) << 1;  // 0 or 2
  // B 4x16 f32 (mirror): lane&15 -> N, half-wave selects K pair
  const int bcol = nbase + (lane & 15);

  const float* __restrict__ hrow = H + (size_t)arow * FIN + khalf;

  v8f c = {};
#pragma unroll
  for (int k0 = 0; k0 < FIN; k0 += 4) {
    v2f a = *(const v2f*)(hrow + k0);  // 8B-aligned: khalf even, FIN even
    v2f b;
    b.x = W[(size_t)(k0 + khalf) * Fout + bcol];
    b.y = W[(size_t)(k0 + khalf + 1) * Fout + bcol];
    c = __builtin_amdgcn_wmma_f32_16x16x4_f32(false, a, false, b,
                                              (short)0, c, false, false);
  }

  // C/D 16x16 f32: VGPR j, lanes 0-15 -> M=j; lanes 16-31 -> M=j+8; N=lane&15
  const int rbase = mbase + ((lane >> 4) << 3);
  const float bv = BIASRELU ? bias[bcol] : 0.0f;
#pragma unroll
  for (int j = 0; j < 8; ++j) {
    size_t idx = (size_t)(rbase + j) * Fout + bcol;
    float v = c[j];
    if constexpr (BETA) v += acc[idx];
    if constexpr (BIASRELU) {
      v += bv;
      v = (v > 0.0f) ? v : NEG_SLOPE * v;
    }
    acc[idx] = v;
  }
}

// ---------------------------------------------------------------------------
// One propagation hop: hout[dst[e],:] += ew[e] * hin[src[e],:]
// One thread per 4 features: b128 gather + 4x global_atomic_add_f32.
// Lane-consecutive float4 chunks -> coalesced gathers/atomics (L2-resident).
// ---------------------------------------------------------------------------
template <int F>
__global__ void scatter_hop(const long long* __restrict__ src,
                            const long long* __restrict__ dst,
                            const float* __restrict__ ew,
                            const float* __restrict__ hin,
                            float* __restrict__ hout, int E) {
  constexpr int TPE = F / 4;  // threads per edge
  long long tid = (long long)blockIdx.x * blockDim.x + threadIdx.x;
  if (tid >= (long long)E * TPE) return;
  int e = (int)(tid / TPE);
  int f4 = (int)(tid % TPE) << 2;
  long long s = src[e], d = dst[e];
  float w = ew[e];
  const float4 hv = *(const float4*)(hin + s * F + f4);  // global_load_b128
  float* base = hout + d * F + f4;
  atomic_fadd_hw(base + 0, w * hv.x);
  atomic_fadd_hw(base + 1, w * hv.y);
  atomic_fadd_hw(base + 2, w * hv.z);
  atomic_fadd_hw(base + 3, w * hv.w);
}

// ---------------------------------------------------------------------------
// Final layer term: out[n] (+)= H[n,:] . w[:]  (Fin=64, Fout=1)
// One wave per node, shuffle reduction.
// ---------------------------------------------------------------------------
template <bool BETA>
__global__ void dot64(const float* __restrict__ H, const float* __restrict__ w,
                      float* __restrict__ out, int nNodes) {
  int node = blockIdx.x * (blockDim.x >> 5) + (threadIdx.x >> 5);
  int lane = threadIdx.x & 31;
  if (node >= nNodes) return;
  const float* h = H + (size_t)node * 64;
  float v = h[lane] * w[lane] + h[lane + 32] * w[lane + 32];
#pragma unroll
  for (int off = 16; off; off >>= 1) v += __shfl_xor(v, off, 32);
  if (lane == 0) out[node] = BETA ? out[node] + v : v;
}

// ---------------------------------------------------------------------------
extern "C" void kernel_launch(void* const* d_in, const int* in_sizes, int n_in,
                              void* d_out, int out_size, void* d_ws,
                              size_t ws_size, hipStream_t stream) {
  const float*     x  = (const float*)d_in[0];      // (N, 32)
  const long long* ei = (const long long*)d_in[1];  // (2, E) int64
  const float*     ew = (const float*)d_in[2];      // (E,)
  const float*     W0 = (const float*)d_in[3];      // (4, 32, 64)
  const float*     b0 = (const float*)d_in[4];      // (64,)
  const float*     W1 = (const float*)d_in[5];      // (4, 64, 64)
  const float*     b1 = (const float*)d_in[6];      // (64,)
  const float*     W2 = (const float*)d_in[7];      // (4, 64, 1)

  const int N = in_sizes[0] / 32;
  const int E = in_sizes[2];
  const long long* src = ei;
  const long long* dst = ei + E;

  // Workspace: three N x 64 f32 buffers (38.4 MB total)
  float* P = (float*)d_ws;
  float* Q = P + (size_t)N * 64;
  float* R = Q + (size_t)N * 64;
  float* out = (float*)d_out;

  const int TPB = 256;  // 8 waves per workgroup
  auto gemm_blocks = [&](int fout) {
    int waves = (N >> 4) * (fout >> 4);
    return (waves + 7) / 8;
  };
  auto sc_blocks = [&](int F) {
    long long t = (long long)E * (F / 4);
    return (int)((t + TPB - 1) / TPB);
  };
  const int dotBlocks = (N + 7) / 8;

  // ============ Layer 0: Fin=32, Fout=64, bias+lrelu, acc in R ============
  tag_gemm_wmma<32, false, false><<<gemm_blocks(64), TPB, 0, stream>>>(x, W0 + 0 * 32 * 64, R, nullptr, N, 64);
  hipMemsetAsync(P, 0, (size_t)N * 32 * 4, stream);
  scatter_hop<32><<<sc_blocks(32), TPB, 0, stream>>>(src, dst, ew, x, P, E);
  tag_gemm_wmma<32, true, false><<<gemm_blocks(64), TPB, 0, stream>>>(P, W0 + 1 * 32 * 64, R, nullptr, N, 64);
  hipMemsetAsync(Q, 0, (size_t)N * 32 * 4, stream);
  scatter_hop<32><<<sc_blocks(32), TPB, 0, stream>>>(src, dst, ew, P, Q, E);
  tag_gemm_wmma<32, true, false><<<gemm_blocks(64), TPB, 0, stream>>>(Q, W0 + 2 * 32 * 64, R, nullptr, N, 64);
  hipMemsetAsync(P, 0, (size_t)N * 32 * 4, stream);
  scatter_hop<32><<<sc_blocks(32), TPB, 0, stream>>>(src, dst, ew, Q, P, E);
  tag_gemm_wmma<32, true, true><<<gemm_blocks(64), TPB, 0, stream>>>(P, W0 + 3 * 32 * 64, R, b0, N, 64);
  // R = leaky_relu(TAGConv0(x))

  // ============ Layer 1: Fin=64, Fout=64, bias+lrelu, acc in Q ============
  tag_gemm_wmma<64, false, false><<<gemm_blocks(64), TPB, 0, stream>>>(R, W1 + 0 * 64 * 64, Q, nullptr, N, 64);
  hipMemsetAsync(P, 0, (size_t)N * 64 * 4, stream);
  scatter_hop<64><<<sc_blocks(64), TPB, 0, stream>>>(src, dst, ew, R, P, E);
  tag_gemm_wmma<64, true, false><<<gemm_blocks(64), TPB, 0, stream>>>(P, W1 + 1 * 64 * 64, Q, nullptr, N, 64);
  hipMemsetAsync(R, 0, (size_t)N * 64 * 4, stream);  // R free after hop 1
  scatter_hop<64><<<sc_blocks(64), TPB, 0, stream>>>(src, dst, ew, P, R, E);
  tag_gemm_wmma<64, true, false><<<gemm_blocks(64), TPB, 0, stream>>>(R, W1 + 2 * 64 * 64, Q, nullptr, N, 64);
  hipMemsetAsync(P, 0, (size_t)N * 64 * 4, stream);
  scatter_hop<64><<<sc_blocks(64), TPB, 0, stream>>>(src, dst, ew, R, P, E);
  tag_gemm_wmma<64, true, true><<<gemm_blocks(64), TPB, 0, stream>>>(P, W1 + 3 * 64 * 64, Q, b1, N, 64);
  // Q = leaky_relu(TAGConv1(...))

  // ============ Layer 2: Fin=64, Fout=1, no bias/act, acc in d_out ============
  dot64<false><<<dotBlocks, TPB, 0, stream>>>(Q, W2 + 0 * 64, out, N);
  hipMemsetAsync(P, 0, (size_t)N * 64 * 4, stream);
  scatter_hop<64><<<sc_blocks(64), TPB, 0, stream>>>(src, dst, ew, Q, P, E);
  dot64<true><<<dotBlocks, TPB, 0, stream>>>(P, W2 + 1 * 64, out, N);
  hipMemsetAsync(R, 0, (size_t)N * 64 * 4, stream);
  scatter_hop<64><<<sc_blocks(64), TPB, 0, stream>>>(src, dst, ew, P, R, E);
  dot64<true><<<dotBlocks, TPB, 0, stream>>>(R, W2 + 2 * 64, out, N);
  hipMemsetAsync(P, 0, (size_t)N * 64 * 4, stream);
  scatter_hop<64><<<sc_blocks(64), TPB, 0, stream>>>(src, dst, ew, R, P, E);
  dot64<true><<<dotBlocks, TPB, 0, stream>>>(P, W2 + 3 * 64, out, N);
}